// pythonDTW_FULL_ROW_38448547233961
// MI455X (gfx1250) — compile-verified
//
#include <hip/hip_runtime.h>

// DTW: input N=65536, kernel K=512. Sequential over rows; one wave32 does the
// row scans (16 cols/lane + 5-step shuffle scans); a second wave produces the
// (k-x)^2 distance rows via V_WMMA_F32_16X16X4_F32 into double-buffered LDS.

#define K_LEN 512
#define N_LEN 65536
#define ROWS_PER_BLK 16
#define NBLK (N_LEN / ROWS_PER_BLK)            // 4096
#define D_STRIDE 516                           // floats/row: 8 rows => +32 banks
#define D_BUF_FLOATS (ROWS_PER_BLK * D_STRIDE) // 8256 (16B aligned: 516*4=2064)
#define SMEM_K_OFF (2 * D_BUF_FLOATS)
#define SMEM_K2_OFF (SMEM_K_OFF + K_LEN)
#define SMEM_FLOATS (SMEM_K2_OFF + K_LEN)      // 17536 floats = 70144 B

typedef float v2f __attribute__((ext_vector_type(2)));
typedef float v8f __attribute__((ext_vector_type(8)));

// 16B-chunk bank swizzle: bijective, spreads stride-4-chunk access 2-way.
__device__ __forceinline__ int swz16(int c) { return c ^ ((c >> 3) & 0xF); }

// Producer: build d[rr][n] = (k[n] - x[blk*16+rr])^2 for one 16-row block
// via WMMA outer expansion: A row m = [x^2, -2x, 1, 0], B col n = [1, k, k^2, 0].
__device__ __forceinline__ void produce_block(int blk, float* __restrict__ sm,
                                              const float* __restrict__ x,
                                              int lane) {
  float* dbuf = sm + (blk & 1) * D_BUF_FLOATS;
  const int lo = lane & 15;
  const bool lowHalf = lane < 16;
  float xm = x[blk * ROWS_PER_BLK + lo];
  v2f a;                       // A 16x4 f32: lanes0-15 K=0,1 ; lanes16-31 K=2,3
  a.x = lowHalf ? xm * xm : 1.0f;
  a.y = lowHalf ? -2.0f * xm : 0.0f;
  const int rowbase = (lowHalf ? 0 : 8) * D_STRIDE;
#pragma unroll 4
  for (int tile = 0; tile < K_LEN / 16; ++tile) {
    int n = tile * 16 + lo;
    float kv = sm[SMEM_K_OFF + n];
    float kv2 = sm[SMEM_K2_OFF + n];
    v2f b;                     // B 4x16 f32: lanes0-15 K=0,1 ; lanes16-31 K=2,3
    b.x = lowHalf ? 1.0f : kv2;
    b.y = lowHalf ? kv : 0.0f;
    v8f acc = {};
    acc = __builtin_amdgcn_wmma_f32_16x16x4_f32(false, a, false, b,
                                                (short)0, acc, false, false);
    int physf = swz16(n >> 2) * 4 + (n & 3);
    // D layout: VGPR r -> row r (lanes 0-15) / row r+8 (lanes 16-31), col=lane&15
#pragma unroll
    for (int r = 0; r < 8; ++r)
      dbuf[rowbase + r * D_STRIDE + physf] = acc[r];
  }
}

// Consumer: 16 sequential DTW row updates. Lane L owns columns [16L, 16L+16).
__device__ __forceinline__ void consume_block(int blk, const float* __restrict__ sm,
                                              float* __restrict__ out,
                                              float prev[16], int lane) {
  const float* dbuf = sm + (blk & 1) * D_BUF_FLOATS;
  const float INF = __builtin_inff();
  for (int rr = 0; rr < ROWS_PER_BLK; ++rr) {
    const int row = blk * ROWS_PER_BLK + rr;
    float dl[16];
#pragma unroll
    for (int j = 0; j < 4; ++j) {
      int chunk = lane * 4 + j;
      const float4 q =
          *(const float4*)(dbuf + rr * D_STRIDE + swz16(chunk) * 4);
      dl[4 * j + 0] = q.x; dl[4 * j + 1] = q.y;
      dl[4 * j + 2] = q.z; dl[4 * j + 3] = q.w;
    }
    // S = inclusive cumsum(d): local serial + wave32 exclusive add-scan
    float s[16];
    s[0] = dl[0];
#pragma unroll
    for (int j = 1; j < 16; ++j) s[j] = s[j - 1] + dl[j];
    float tv = s[15];
#pragma unroll
    for (int off = 1; off < 32; off <<= 1) {   // wave32 (gfx1250)
      float t = __shfl_up(tv, off);
      if (lane >= off) tv += t;
    }
    float addoff = __shfl_up(tv, 1);
    if (lane == 0) addoff = 0.0f;
    float S[16];
#pragma unroll
    for (int j = 0; j < 16; ++j) S[j] = s[j] + addoff;

    float ac[16];
    if (row == 0) {
#pragma unroll
      for (int j = 0; j < 16; ++j) ac[j] = S[j];   // row 0: ac = cumsum(d)
    } else {
      float pedge = __shfl_up(prev[15], 1);        // prev[-1] from lane-1
      if (lane == 0) pedge = INF;                  // prev_shift[0] = +inf
      float z[16];
      z[0] = fminf(pedge, prev[0]) + dl[0] - S[0];
#pragma unroll
      for (int j = 1; j < 16; ++j)
        z[j] = fminf(prev[j - 1], prev[j]) + dl[j] - S[j];
      // inclusive cummin(z): local serial + wave32 exclusive min-scan
      float m[16];
      m[0] = z[0];
#pragma unroll
      for (int j = 1; j < 16; ++j) m[j] = fminf(m[j - 1], z[j]);
      float mv = m[15];
#pragma unroll
      for (int off = 1; off < 32; off <<= 1) {
        float t = __shfl_up(mv, off);
        if (lane >= off) mv = fminf(mv, t);
      }
      float minoff = __shfl_up(mv, 1);
      if (lane == 0) minoff = INF;
#pragma unroll
      for (int j = 0; j < 16; ++j) ac[j] = S[j] + fminf(m[j], minoff);
    }
#pragma unroll
    for (int j = 0; j < 16; ++j) prev[j] = ac[j];
    if (lane == 31) out[row] = ac[15];             // last column
  }
}

extern "C" __global__ void __launch_bounds__(64, 1)
dtw_full_row_kernel(const float* __restrict__ x, const float* __restrict__ k,
                    float* __restrict__ out) {
  extern __shared__ float sm[];
  const int tid = threadIdx.x;
  const int lane = tid & 31;
  const int wid = tid >> 5;

  // Stage kernel & kernel^2 into LDS once.
  for (int i = tid; i < K_LEN; i += 64) {
    float kv = k[i];
    sm[SMEM_K_OFF + i] = kv;
    sm[SMEM_K2_OFF + i] = kv * kv;
  }
  __syncthreads();

  if (wid == 0) produce_block(0, sm, x, lane);     // prologue fill of buf0
  __syncthreads();

  float prev[16];
#pragma unroll
  for (int j = 0; j < 16; ++j) prev[j] = 0.0f;

  for (int blk = 0; blk < NBLK; ++blk) {
    if (wid == 0) {
      if (blk + 1 < NBLK) produce_block(blk + 1, sm, x, lane);
    } else {
      consume_block(blk, sm, out, prev, lane);
    }
    __syncthreads();  // buf[(blk+1)&1] ready; buf[blk&1] free for reuse
  }
}

extern "C" void kernel_launch(void* const* d_in, const int* in_sizes, int n_in,
                              void* d_out, int out_size, void* d_ws, size_t ws_size,
                              hipStream_t stream) {
  (void)in_sizes; (void)n_in; (void)d_ws; (void)ws_size; (void)out_size;
  const float* x = (const float*)d_in[0];   // input, 65536 f32
  const float* k = (const float*)d_in[1];   // kernel, 512 f32
  float* out = (float*)d_out;               // 65536 f32
  hipLaunchKernelGGL(dtw_full_row_kernel, dim3(1), dim3(64),
                     SMEM_FLOATS * sizeof(float), stream, x, k, out);
}